// RollerPooling_58463094833682
// MI455X (gfx1250) — compile-verified
//
#include <hip/hip_runtime.h>

// RollerPooling forward for MI455X (gfx1250).
// Memory-bound (514 MB @ 23.3 TB/s => ~22us floor). Strategy: wave-per-edge,
// async global->LDS double-buffered row fetch (ASYNCcnt pipelining), wave32
// shuffle scan, LDS gather of windowed prefix sums, non-temporal b128 stores.

typedef __attribute__((ext_vector_type(4))) float v4f;

#define WPB 8  // waves per 256-thread block (wave32)

__device__ __forceinline__ void async_row_load(unsigned long long gbase,
                                               unsigned voff, unsigned ldsoff) {
  // GVS mode: mem = SGPR base + 32-bit VGPR offset; VDST holds LDS byte addr.
  asm volatile("global_load_async_to_lds_b128 %0, %1, %2"
               :: "v"(ldsoff), "v"(voff), "s"(gbase)
               : "memory");
}
__device__ __forceinline__ void wait_async_le1() {
  asm volatile("s_wait_asynccnt 0x1" ::: "memory");
}
__device__ __forceinline__ void wait_async_le0() {
  asm volatile("s_wait_asynccnt 0x0" ::: "memory");
}

__global__ void __launch_bounds__(256)
roller_pool_fwd(const float* __restrict__ hidden,
                const int* __restrict__ tsz,
                float* __restrict__ out, int E) {
  __shared__ __align__(16) float rows[WPB][2][128];  // async landing pads
  __shared__ float csum[WPB][136];                   // 129 used + pad

  const int lane   = threadIdx.x & 31;
  const int wv     = threadIdx.x >> 5;
  const int wave   = blockIdx.x * WPB + wv;
  const int nwaves = gridDim.x * WPB;
  if (wave >= E) return;  // wave-uniform exit

  const unsigned long long gbase = (unsigned long long)hidden;
  const unsigned laneByte = (unsigned)lane * 16u;
  // Generic LDS addresses carry the LDS byte offset in the low 32 bits.
  const unsigned ldsRow0 = (unsigned)(size_t)&rows[wv][0][lane * 4];
  const unsigned ldsRow1 = (unsigned)(size_t)&rows[wv][1][lane * 4];
  float* const cs = &csum[wv][0];
  const int j0 = lane * 4;

  // Prologue: stage first row into buffer 0 (ASYNCcnt = 1).
  async_row_load(gbase, (unsigned)wave * 512u + laneByte, ldsRow0);

  int buf = 0;
  for (int e = wave; e < E; e += nwaves) {
    const bool hasNext = (e + nwaves) < E;
    if (hasNext)  // prefetch next edge's row into the other buffer
      async_row_load(gbase, (unsigned)(e + nwaves) * 512u + laneByte,
                     buf ? ldsRow0 : ldsRow1);

    const int w = 129 - tsz[e];  // window size, in [1,128]

    // Async loads complete in order: cnt<=1 means the current buffer landed.
    if (hasNext) wait_async_le1(); else wait_async_le0();

    const float* rp = &rows[wv][buf][0];
    v4f v = *(const v4f*)(rp + j0);  // ds_load_b128, my 4 channels

    // 4-wide local inclusive prefix
    float p0 = v.x;
    float p1 = p0 + v.y;
    float p2 = p1 + v.z;
    float p3 = p2 + v.w;

    // wave32 inclusive scan of per-lane totals
    float s = p3;
#pragma unroll
    for (int d = 1; d < 32; d <<= 1) {
      float t = __shfl_up(s, d, 32);
      if (lane >= d) s += t;
    }
    const float ex = s - p3;  // exclusive base at channel j0

    // exclusive csum values at channels j0..j0+3 (kept in regs AND LDS)
    const float c0 = ex, c1 = ex + p0, c2 = ex + p1, c3 = ex + p2;
    cs[j0 + 0] = c0; cs[j0 + 1] = c1; cs[j0 + 2] = c2; cs[j0 + 3] = c3;
    if (lane == 31) cs[128] = s;  // total sum = csum[n]
    // Same-wave DS ops are in-order: stores above are visible to loads below.

    const float invw = 1.0f / (float)w;
    int t0 = j0 + 0 + w; if (t0 > 128) t0 = 128;
    int t1 = j0 + 1 + w; if (t1 > 128) t1 = 128;
    int t2 = j0 + 2 + w; if (t2 > 128) t2 = 128;
    int t3 = j0 + 3 + w; if (t3 > 128) t3 = 128;

    v4f o;
    o.x = (cs[t0] - c0) * invw;
    o.y = (cs[t1] - c1) * invw;
    o.z = (cs[t2] - c2) * invw;
    o.w = (cs[t3] - c3) * invw;

    // 512 MB streamed through a 192 MB L2: non-temporal b128 store.
    __builtin_nontemporal_store(o, (v4f*)(out + (size_t)e * 128 + j0));

    buf ^= 1;
  }
}

extern "C" void kernel_launch(void* const* d_in, const int* in_sizes, int n_in,
                              void* d_out, int out_size, void* d_ws, size_t ws_size,
                              hipStream_t stream) {
  (void)n_in; (void)out_size; (void)d_ws; (void)ws_size;
  const float* hidden = (const float*)d_in[0];
  const int*   tsz    = (const int*)d_in[1];
  float*       out    = (float*)d_out;
  const int E = in_sizes[1];  // N_EDGES (target_size element count)

  int blocks = (E + WPB - 1) / WPB;
  if (blocks > 4096) blocks = 4096;  // ~32K waves: fills a big CDNA5 part,
                                     // still ~15 pipelined rows per wave
  if (blocks < 1) blocks = 1;
  roller_pool_fwd<<<blocks, 256, 0, stream>>>(hidden, tsz, out, E);
}